// PANormSelective2d_15006615732764
// MI455X (gfx1250) — compile-verified
//
#include <hip/hip_runtime.h>
#include <hip/hip_bf16.h>

// PANormSelective2d for MI455X (gfx1250, wave32).
// Shapes fixed by the reference: N=32, C=256, H=W=56 (HW=3136), GROUPS=8.
// Strategy:
//   K1: per-(n,c) sum/sumsq (reads x once from HBM, warms the 192MB L2)
//   K2: S2 = sum over positions of (sum over channels x)^2, computed with
//       V_WMMA_F32_16X16X4_F32 (A = ones(16x4), B = 4 channels x 16 positions).
//       Loads are unconditional (addresses clamped for the few fully-OOB
//       waves) so no EXEC juggling / per-load waits — contribution is masked
//       at the squaring step instead.
//   K3: tiny finalize -> gates -> per-(n,c) affine coefficients A,B
//   K4: y = A*x + B with non-temporal stores (keep x resident in L2)

#define N_    32
#define C_    256
#define HW_   3136          // 56*56 = 196*16 (tiles never straddle)
#define NHW_  100352.0f     // 32*3136
#define CHW_  802816.0f     // 256*3136
#define GHW_  100352.0f     // 32*3136 (channels-per-group * HW)
#define EPS_  1e-5f
#define K2_TILES 25         // ceil(196 wave-tiles / 8 waves)

typedef float v2f __attribute__((ext_vector_type(2)));
typedef float v4f __attribute__((ext_vector_type(4)));
typedef float v8f __attribute__((ext_vector_type(8)));

// ---------------------------------------------------------------- K1
__global__ void k_plane_stats(const float* __restrict__ x,
                              float* __restrict__ tsum,
                              float* __restrict__ tsq) {
  const int c = blockIdx.x, n = blockIdx.y;
  const int plane = n * C_ + c;
  const v4f* xv = (const v4f*)(x + (size_t)plane * HW_);
  float s = 0.0f, q = 0.0f;
  for (int i = threadIdx.x; i < HW_ / 4; i += 256) {
    v4f v = xv[i];
    s += v.x + v.y + v.z + v.w;
    q += v.x * v.x + v.y * v.y + v.z * v.z + v.w * v.w;
  }
  __shared__ float shs[256], shq[256];
  const int tid = threadIdx.x;
  shs[tid] = s; shq[tid] = q;
  __syncthreads();
  for (int off = 128; off > 0; off >>= 1) {
    if (tid < off) { shs[tid] += shs[tid + off]; shq[tid] += shq[tid + off]; }
    __syncthreads();
  }
  if (tid == 0) { tsum[plane] = shs[0]; tsq[plane] = shq[0]; }
}

// ---------------------------------------------------------------- K2
// Each wave owns 16 spatial positions; loops channels 4 at a time, using the
// matrix pipe to accumulate per-position channel sums:
//   D = ones(16x4) x B(4x16) + C  =>  D[m][n] += sum_k B[k][n]
// After 64 iterations acc[0] (row M=0 or 8, col N=lane&15) is the full
// 256-channel sum for that position. Square + reduce -> per-block partial.
__global__ void k_chansum_sq(const float* __restrict__ x,
                             float* __restrict__ part) {
  const int n     = blockIdx.y;
  const int wave  = threadIdx.x >> 5;
  const int lane  = threadIdx.x & 31;
  const int tile  = blockIdx.x * 8 + wave;        // 16-position tile id
  const int p     = tile * 16 + (lane & 15);
  const int khalf = lane >> 4;
  const bool inb  = (p < HW_);                    // uniform per wave (3136%16==0)
  // Clamp instead of predicating: OOB waves load valid-but-unused data, so
  // the loads stay unconditional (no exec-mask branches, loads can batch).
  const int psafe = inb ? p : (lane & 15);
  const float* xb = x + (size_t)n * C_ * HW_ + psafe;

  v2f a; a[0] = 1.0f; a[1] = 1.0f;                // A = ones(16x4)
  v8f acc = {};
#pragma unroll 4
  for (int c = 0; c < C_; c += 4) {
    const int ch = c + 2 * khalf;
    v2f b;
    b[0] = xb[(size_t)ch * HW_];
    b[1] = xb[(size_t)(ch + 1) * HW_];
    acc = __builtin_amdgcn_wmma_f32_16x16x4_f32(
        /*neg_a=*/false, a, /*neg_b=*/false, b,
        /*c_mod=*/(short)0, acc, /*reuse_a=*/false, /*reuse_b=*/false);
  }
  float s   = acc[0];
  float val = (inb && khalf == 0) ? s * s : 0.0f; // count each position once
  for (int off = 16; off > 0; off >>= 1) val += __shfl_down(val, off, 32);

  __shared__ float wsum[8];
  if (lane == 0) wsum[wave] = val;
  __syncthreads();
  if (threadIdx.x == 0) {
    float t = 0.0f;
    for (int w2 = 0; w2 < 8; ++w2) t += wsum[w2];
    part[blockIdx.y * gridDim.x + blockIdx.x] = t;
  }
}

// ---------------------------------------------------------------- K3
__global__ void k_finalize(const float* __restrict__ tsum,
                           const float* __restrict__ tsq,
                           const float* __restrict__ part,   // 32*K2_TILES
                           const float* __restrict__ weight,
                           const float* __restrict__ bias,
                           const float* __restrict__ gate_logits,
                           const float* __restrict__ diag_proj,
                           float* __restrict__ Ac, float* __restrict__ Bc) {
  const int tid = threadIdx.x;
  __shared__ float mu_b[C_], iv_b[C_];
  __shared__ float mu_l[N_], iv_l[N_], varl[N_];
  __shared__ float mu_g[C_], iv_g[C_];
  __shared__ float red[256];
  __shared__ float scal[8];   // 0:bv_mean 2:total_sumsq 3:S2 4..6:gates

  // per-channel (batch norm) stats; keep per-channel sumsq for total
  float qc;
  {
    const int c = tid;
    float s = 0.0f, q = 0.0f;
    for (int n = 0; n < N_; ++n) { s += tsum[n * C_ + c]; q += tsq[n * C_ + c]; }
    qc = q;
    float m = s / NHW_;
    float v = q / NHW_ - m * m;
    mu_b[c] = m;
    iv_b[c] = 1.0f / sqrtf(v + EPS_);
    red[tid] = v;
  }
  __syncthreads();
  for (int off = 128; off > 0; off >>= 1) {
    if (tid < off) red[tid] += red[tid + off];
    __syncthreads();
  }
  if (tid == 0) scal[0] = red[0] / (float)C_;     // batch_var mean
  __syncthreads();
  red[tid] = qc;
  __syncthreads();
  for (int off = 128; off > 0; off >>= 1) {
    if (tid < off) red[tid] += red[tid + off];
    __syncthreads();
  }
  if (tid == 0) scal[2] = red[0];                 // total sum of x^2
  __syncthreads();
  {
    float s = 0.0f;
    for (int i = tid; i < N_ * K2_TILES; i += 256) s += part[i];
    red[tid] = s;
  }
  __syncthreads();
  for (int off = 128; off > 0; off >>= 1) {
    if (tid < off) red[tid] += red[tid + off];
    __syncthreads();
  }
  if (tid == 0) scal[3] = red[0];                 // S2
  __syncthreads();

  // per-sample (layer norm) stats
  if (tid < N_) {
    float s = 0.0f, q = 0.0f;
    for (int c = 0; c < C_; ++c) { s += tsum[tid * C_ + c]; q += tsq[tid * C_ + c]; }
    float m = s / CHW_;
    float v = q / CHW_ - m * m;
    mu_l[tid] = m; iv_l[tid] = 1.0f / sqrtf(v + EPS_); varl[tid] = v;
  }
  // per-(sample,group) stats: tid = n*8+g
  {
    const int n = tid >> 3, g = tid & 7;
    float s = 0.0f, q = 0.0f;
    for (int c = g * 32; c < g * 32 + 32; ++c) { s += tsum[n * C_ + c]; q += tsq[n * C_ + c]; }
    float m = s / GHW_;
    float v = q / GHW_ - m * m;
    mu_g[tid] = m; iv_g[tid] = 1.0f / sqrtf(v + EPS_);
  }
  __syncthreads();

  if (tid == 0) {
    float sv = 0.0f;
    for (int n = 0; n < N_; ++n) sv += varl[n];
    sv /= (float)N_;                                       // sample_var mean
    float bv  = scal[0];
    float msq = scal[2] / (NHW_ * (float)C_);              // mean of x^2
    float cv  = msq - scal[3] / (NHW_ * (float)C_ * (float)C_);  // channel_var
    float dv[3] = { bv, sv, cv };
    float d[3];
    for (int i = 0; i < 3; ++i) d[i] = tanhf(logf(fmaxf(dv[i], EPS_)));
    float l[3], mx = -1e30f;
    for (int i = 0; i < 3; ++i) {
      l[i] = gate_logits[i];
      for (int j = 0; j < 3; ++j) l[i] += diag_proj[i * 3 + j] * d[j];
      mx = fmaxf(mx, l[i]);
    }
    float se = 0.0f;
    for (int i = 0; i < 3; ++i) { l[i] = expf(l[i] - mx); se += l[i]; }
    for (int i = 0; i < 3; ++i) scal[4 + i] = l[i] / se;
  }
  __syncthreads();

  const float g0 = scal[4], g1 = scal[5], g2 = scal[6];
  for (int idx = tid; idx < N_ * C_; idx += 256) {
    const int n = idx >> 8, c = idx & 255, gr = c >> 5;
    const float w  = weight[c];
    const float ib = iv_b[c], il = iv_l[n], ig = iv_g[n * 8 + gr];
    const float A  = w * (g0 * ib + g1 * il + g2 * ig);
    const float Bv = bias[c] - w * (g0 * mu_b[c] * ib + g1 * mu_l[n] * il +
                                    g2 * mu_g[n * 8 + gr] * ig);
    Ac[idx] = A; Bc[idx] = Bv;
  }
}

// ---------------------------------------------------------------- K4
__global__ void k_apply(const float* __restrict__ x,
                        const float* __restrict__ Ac,
                        const float* __restrict__ Bc,
                        float* __restrict__ y) {
  const int c = blockIdx.x, n = blockIdx.y;
  const int idx = n * C_ + c;
  const float a = Ac[idx], b = Bc[idx];
  const size_t off = (size_t)idx * HW_;
  const v4f* xv = (const v4f*)(x + off);
  v4f* yv = (v4f*)(y + off);
  for (int i = threadIdx.x; i < HW_ / 4; i += 256) {
    v4f v = xv[i];                       // RT load: hits L2 (x is resident)
    v4f r = v * a + b;
    __builtin_nontemporal_store(r, yv + i);  // NT store: don't evict x
  }
}

// ---------------------------------------------------------------- launch
extern "C" void kernel_launch(void* const* d_in, const int* in_sizes, int n_in,
                              void* d_out, int out_size, void* d_ws, size_t ws_size,
                              hipStream_t stream) {
  const float* x           = (const float*)d_in[0];
  const float* weight      = (const float*)d_in[1];
  const float* bias        = (const float*)d_in[2];
  const float* gate_logits = (const float*)d_in[3];
  const float* diag_proj   = (const float*)d_in[4];
  float* y  = (float*)d_out;
  float* ws = (float*)d_ws;

  float* tsum = ws;                 // 8192 floats
  float* tsq  = ws + 8192;          // 8192 floats
  float* Ac   = ws + 16384;         // 8192 floats
  float* Bc   = ws + 24576;         // 8192 floats
  float* part = ws + 32768;         // 32*K2_TILES = 800 floats

  k_plane_stats<<<dim3(C_, N_), 256, 0, stream>>>(x, tsum, tsq);
  k_chansum_sq<<<dim3(K2_TILES, N_), 256, 0, stream>>>(x, part);
  k_finalize<<<1, 256, 0, stream>>>(tsum, tsq, part, weight, bias,
                                    gate_logits, diag_proj, Ac, Bc);
  k_apply<<<dim3(C_, N_), 256, 0, stream>>>(x, Ac, Bc, y);
}